// NeiAttention_85409719648982
// MI455X (gfx1250) — compile-verified
//
#include <hip/hip_runtime.h>

typedef __attribute__((ext_vector_type(2))) float v2f;
typedef __attribute__((ext_vector_type(8))) float v8f;

#define LDSW_STRIDE 196   // 192 + 4 pad: 196 % 64 == 4 -> conflict-free b64 B-frag reads
#define WAVES_PER_BLOCK 8
#define PAIRS_PER_WAVE 2  // register blocking: one B-frag LDS read feeds 2 WMMAs

// One wave32 per TWO (batch, node) pairs:
//   V_tile[2][16 x 128] = Xnei[32 x 192] @ W^T[192 x 128] + bias  (f32 WMMA, K-steps of 4)
//   alpha = softmax( (x . V rows) / sqrt(128) )                   (in-register, shuffles)
//   out   = alpha^T @ V_tile                                      (in-register)
__global__ __launch_bounds__(256, 1) void nei_attn_wmma_f32(
    const float* __restrict__ x,     // [B,32,128]
    const float* __restrict__ rel,   // [B,512,64]
    const float* __restrict__ node,  // [B,512,128]
    const float* __restrict__ Wm,    // [128,192] row-major
    const float* __restrict__ bias,  // [128]
    float* __restrict__ out)         // [B,32,128]
{
    __shared__ float ldsW[128 * LDSW_STRIDE];  // ~100 KB: 3 WGs / WGP LDS-wise

    const int tid = threadIdx.x;

    // ---- stage W (128x192) into bank-padded LDS, float2 granularity ----
    {
        const v2f* Wv = (const v2f*)Wm;            // 128*96 float2
        #pragma unroll 4
        for (int i = tid; i < 128 * 96; i += 256) {
            int row = i / 96;
            int c2  = i - row * 96;
            *(v2f*)&ldsW[row * LDSW_STRIDE + 2 * c2] = Wv[i];
        }
    }
    __syncthreads();

    const int wave = tid >> 5;
    const int lane = tid & 31;
    const int lo   = lane & 15;   // M row (A) / N col (B,C,D)
    const int hi   = lane >> 4;   // K sub-offset select

    const int pair0 = (blockIdx.x * WAVES_PER_BLOCK + wave) * PAIRS_PER_WAVE; // b*32 + n
    // A-matrix rows: neighbor rows pair*16 .. pair*16+15; this lane owns row +lo.
    // Pair1's rows are the next contiguous 16 rows.
    const long r0 = (long)pair0 * 16 + lo;
    const float* relRow0  = rel  + r0 * 64  + 2 * hi;
    const float* relRow1  = relRow0 + 16 * 64;
    const float* nodeRow0 = node + r0 * 128 + 2 * hi;
    const float* nodeRow1 = nodeRow0 + 16 * 128;

    // ---- accumulators: 2 pairs x 8 N-tiles of 16 features; init with bias[d] ----
    // C/D layout: acc[p][t][j] = V_p[s = j + 8*hi][d = 16*t + lo]
    v8f acc[PAIRS_PER_WAVE][8];
    #pragma unroll
    for (int t = 0; t < 8; ++t) {
        float bv = bias[16 * t + lo];
        v8f b8 = (v8f){bv, bv, bv, bv, bv, bv, bv, bv};
        acc[0][t] = b8;
        acc[1][t] = b8;
    }

    // ---- K-loop over edge features (k = 0..63) ----
    #pragma unroll 4
    for (int kk = 0; kk < 16; ++kk) {
        const int k0 = kk * 4;
        v2f a0 = *(const v2f*)(relRow0 + k0);      // A frag: K = k0+2*hi, +1
        v2f a1 = *(const v2f*)(relRow1 + k0);
        #pragma unroll
        for (int t = 0; t < 8; ++t) {
            v2f bfrag = *(const v2f*)&ldsW[(16 * t + lo) * LDSW_STRIDE + k0 + 2 * hi];
            acc[0][t] = __builtin_amdgcn_wmma_f32_16x16x4_f32(
                false, a0, false, bfrag, (short)0, acc[0][t], false, false);
            acc[1][t] = __builtin_amdgcn_wmma_f32_16x16x4_f32(
                false, a1, false, bfrag, (short)0, acc[1][t], false, false);
        }
    }
    // ---- K-loop over node features (k = 64..191) ----
    #pragma unroll 4
    for (int kk = 0; kk < 32; ++kk) {
        const int k0 = kk * 4;
        v2f a0 = *(const v2f*)(nodeRow0 + k0);
        v2f a1 = *(const v2f*)(nodeRow1 + k0);
        #pragma unroll
        for (int t = 0; t < 8; ++t) {
            v2f bfrag = *(const v2f*)&ldsW[(16 * t + lo) * LDSW_STRIDE + 64 + k0 + 2 * hi];
            acc[0][t] = __builtin_amdgcn_wmma_f32_16x16x4_f32(
                false, a0, false, bfrag, (short)0, acc[0][t], false, false);
            acc[1][t] = __builtin_amdgcn_wmma_f32_16x16x4_f32(
                false, a1, false, bfrag, (short)0, acc[1][t], false, false);
        }
    }

    // ---- attention epilogue, per pair, all in-register ----
    const float norm = 0.08838834764831843f;      // 1/sqrt(128)
    #pragma unroll
    for (int pp = 0; pp < PAIRS_PER_WAVE; ++pp) {
        const int pair = pair0 + pp;
        const float* xrow = x + (long)pair * 128;
        float xv[8];
        #pragma unroll
        for (int t = 0; t < 8; ++t) xv[t] = xrow[16 * t + lo];

        // logits: p[j] = partial of alpha[s = j + 8*hi] over this lane's 8 features
        float p[8];
        #pragma unroll
        for (int j = 0; j < 8; ++j) {
            float s = 0.f;
            #pragma unroll
            for (int t = 0; t < 8; ++t) s += acc[pp][t][j] * xv[t];
            p[j] = s;
        }
        // reduce over the 16 lanes of each half (xor 1,2,4,8 stays inside the half)
        #pragma unroll
        for (int m = 1; m <= 8; m <<= 1) {
            #pragma unroll
            for (int j = 0; j < 8; ++j) p[j] += __shfl_xor(p[j], m, 32);
        }

        // softmax over the 16 neighbors; each lane keeps only its half's 8 weights
        #pragma unroll
        for (int j = 0; j < 8; ++j) p[j] *= norm;

        float mx = p[0];
        #pragma unroll
        for (int j = 1; j < 8; ++j) mx = fmaxf(mx, p[j]);
        mx = fmaxf(mx, __shfl_xor(mx, 16, 32));   // global max over all 16 s

        float e[8];
        float z = 0.f;
        #pragma unroll
        for (int j = 0; j < 8; ++j) { e[j] = expf(p[j] - mx); z += e[j]; }
        z += __shfl_xor(z, 16, 32);               // global denominator
        const float invz = 1.0f / z;
        #pragma unroll
        for (int j = 0; j < 8; ++j) e[j] *= invz; // weight for s = j + 8*hi

        // weighted neighbor sum: out[d] = sum_s w[s] * V[s,d]
        float* orow = out + (long)pair * 128;
        #pragma unroll
        for (int t = 0; t < 8; ++t) {
            float po = 0.f;
            #pragma unroll
            for (int j = 0; j < 8; ++j) po += e[j] * acc[pp][t][j];
            po += __shfl_xor(po, 16, 32);         // lane l and l^16 share feature d
            // lanes hi==0 store tiles 0..3, hi==1 store tiles 4..7
            if ((t >> 2) == hi) orow[16 * t + lo] = po;
        }
    }
}

extern "C" void kernel_launch(void* const* d_in, const int* in_sizes, int n_in,
                              void* d_out, int out_size, void* d_ws, size_t ws_size,
                              hipStream_t stream) {
    (void)n_in; (void)out_size; (void)d_ws; (void)ws_size;
    const float* x    = (const float*)d_in[0];
    const float* rel  = (const float*)d_in[1];
    const float* node = (const float*)d_in[2];
    const float* Wm   = (const float*)d_in[3];
    const float* bias = (const float*)d_in[4];
    // d_in[5] is the python scalar i == 0 (S=16), unused.
    float* out = (float*)d_out;

    const int B = in_sizes[0] / (32 * 128);                      // 512
    const int pairs = B * 32;                                    // 16384
    const int blocks = pairs / (WAVES_PER_BLOCK * PAIRS_PER_WAVE); // 1024
    nei_attn_wmma_f32<<<blocks, 256, 0, stream>>>(x, rel, node, Wm, bias, out);
}